// CoRE_55456617726512
// MI455X (gfx1250) — compile-verified
//
#include <hip/hip_runtime.h>
#include <hip/hip_bf16.h>
#include <math.h>

typedef __attribute__((ext_vector_type(2))) float v2f;
typedef __attribute__((ext_vector_type(8))) float v8f;

#define WMMA_F32(a, b, c) \
  __builtin_amdgcn_wmma_f32_16x16x4_f32(false, (a), false, (b), (short)0, (c), false, false)

// 8-byte-aligned vector load of two contiguous floats (lets the backend emit
// global_load_b64 instead of two global_load_b32).
__device__ __forceinline__ v2f ldv2(const float* __restrict__ p) {
  return *(const v2f*)__builtin_assume_aligned(p, 8);
}

// ---------------------------------------------------------------------------
// helpers
// ---------------------------------------------------------------------------
__device__ __forceinline__ void atomicMaxF(float* addr, float v) {
  if (v >= 0.0f) atomicMax((int*)addr, __float_as_int(v));
  else           atomicMin((unsigned int*)addr, __float_as_uint(v));
}

__global__ void fill_kernel(float* __restrict__ p, float v, int n) {
  int i = blockIdx.x * blockDim.x + threadIdx.x;
  if (i < n) p[i] = v;
}

__global__ void reduce_sum_kernel(const float* __restrict__ x, int n, float* __restrict__ out) {
  __shared__ float red[256];
  float s = 0.0f;
  for (int i = blockIdx.x * blockDim.x + threadIdx.x; i < n; i += gridDim.x * blockDim.x) s += x[i];
  red[threadIdx.x] = s;
  __syncthreads();
  for (int w = 128; w > 0; w >>= 1) {
    if ((int)threadIdx.x < w) red[threadIdx.x] += red[threadIdx.x + w];
    __syncthreads();
  }
  if (threadIdx.x == 0) atomicAdd(out, red[0]);
}

__global__ void pad_copy_kernel(const float* __restrict__ src, float* __restrict__ dst,
                                int realElems, int padElems) {
  int i = blockIdx.x * blockDim.x + threadIdx.x;
  if (i < padElems) dst[i] = (i < realElems) ? src[i] : 0.0f;
}

// ---------------------------------------------------------------------------
// Generic f32 WMMA GEMM.  C[M,N] = alpha*op(A,B) (+bias[n]) (+rc*resid[m,n])
// TRANSB=true : B stored [N,K] row-major (C = A @ B^T)
// TRANSB=false: B stored [K,N] row-major (C = A @ B)
// All of M,N multiples of 16; K multiple of 4; all leading dims even.
// ---------------------------------------------------------------------------
template <bool TRANSB>
__global__ void wmma_gemm_kernel(const float* __restrict__ A, const float* __restrict__ B,
                                 float* __restrict__ C, int M, int N, int K,
                                 int lda, int ldb, int ldc, float alphaScale,
                                 const float* __restrict__ bias,
                                 const float* __restrict__ resid, float residCoef) {
  int wave = (int)(((size_t)blockIdx.x * blockDim.x + threadIdx.x) >> 5);
  int lane = threadIdx.x & 31;
  int ntn = N >> 4;
  int total = (M >> 4) * ntn;
  if (wave >= total) return;
  int tm = wave / ntn, tn = wave - tm * ntn;
  int row0 = tm << 4, col0 = tn << 4;
  int half = lane >> 4, lr = lane & 15;
  const float* arow = A + (size_t)(row0 + lr) * lda + 2 * half;
  const float* brow = TRANSB ? (B + (size_t)(col0 + lr) * ldb + 2 * half) : nullptr;
  v8f acc = {};
  for (int k = 0; k < K; k += 4) {
    v2f a = ldv2(arow + k);
    v2f b;
    if (TRANSB) {
      b = ldv2(brow + k);
    } else {
      int ka = k + 2 * half;
      b.x = B[(size_t)ka * ldb + col0 + lr];
      b.y = B[(size_t)(ka + 1) * ldb + col0 + lr];
    }
    acc = WMMA_F32(a, b, acc);
  }
  int col = col0 + lr;
  float bv = bias ? bias[col] : 0.0f;
#pragma unroll
  for (int r = 0; r < 8; r++) {
    int row = row0 + r + 8 * half;
    float v = acc[r] * alphaScale + bv;
    if (resid) v += residCoef * resid[(size_t)row * ldc + col];
    C[(size_t)row * ldc + col] = v;
  }
}

// C[i,j] = alpha * sum_n A[n,i]*B[n,j]   (A:[K,M], B:[K,N] row-major; strided)
__global__ void wmma_gemm_tn_kernel(const float* __restrict__ A, const float* __restrict__ B,
                                    float* __restrict__ C, int M, int N, int K,
                                    int lda, int ldb, int ldc, float alphaScale) {
  int wave = (int)(((size_t)blockIdx.x * blockDim.x + threadIdx.x) >> 5);
  int lane = threadIdx.x & 31;
  int ntn = N >> 4;
  if (wave >= (M >> 4) * ntn) return;
  int tm = wave / ntn, tn = wave - tm * ntn;
  int row0 = tm << 4, col0 = tn << 4;
  int half = lane >> 4, lr = lane & 15;
  v8f acc = {};
  for (int k = 0; k < K; k += 4) {
    int ka = k + 2 * half;
    v2f a, b;
    a.x = A[(size_t)ka * lda + row0 + lr];
    a.y = A[(size_t)(ka + 1) * lda + row0 + lr];
    b.x = B[(size_t)ka * ldb + col0 + lr];
    b.y = B[(size_t)(ka + 1) * ldb + col0 + lr];
    acc = WMMA_F32(a, b, acc);
  }
  int col = col0 + lr;
#pragma unroll
  for (int r = 0; r < 8; r++)
    C[(size_t)(row0 + r + 8 * half) * ldc + col] = acc[r] * alphaScale;
}

// loss += invM * sum_ij (U_i . V_j + 1e-6)^2   over [M,N] with K=inner dim
__global__ void rec_loss_kernel(const float* __restrict__ U, const float* __restrict__ V,
                                float* __restrict__ out, int M, int N, int K, float invM) {
  int wave = (int)(((size_t)blockIdx.x * blockDim.x + threadIdx.x) >> 5);
  int lane = threadIdx.x & 31;
  int ntn = N >> 4;
  if (wave >= (M >> 4) * ntn) return;
  int tm = wave / ntn, tn = wave - tm * ntn;
  int row0 = tm << 4, col0 = tn << 4;
  int half = lane >> 4, lr = lane & 15;
  const float* urow = U + (size_t)(row0 + lr) * K + 2 * half;
  const float* vrow = V + (size_t)(col0 + lr) * K + 2 * half;
  v8f acc = {};
  for (int k = 0; k < K; k += 4) acc = WMMA_F32(ldv2(urow + k), ldv2(vrow + k), acc);
  float local = 0.0f;
#pragma unroll
  for (int r = 0; r < 8; r++) {
    float d = acc[r] + 1e-6f;
    local += d * d;
  }
  for (int off = 16; off >= 1; off >>= 1) local += __shfl_xor(local, off, 32);
  if (lane == 0) atomicAdd(out, local * invM);
}

// cross_align: loss += (1/Mreal)*sum (Gt - Gs + eps)^2; Gs/Gt from unnormalized
// gram tiles scaled by per-row inverse norms.  qs/qt: [Mp,K] row-major (padded).
__global__ void align_loss_kernel(const float* __restrict__ qs, const float* __restrict__ qt,
                                  const float* __restrict__ invns, const float* __restrict__ invnt,
                                  float* __restrict__ out, int Mp, int Mreal, int K) {
  int wave = (int)(((size_t)blockIdx.x * blockDim.x + threadIdx.x) >> 5);
  int lane = threadIdx.x & 31;
  int ntn = Mp >> 4;
  if (wave >= ntn * ntn) return;
  int tm = wave / ntn, tn = wave - tm * ntn;
  int row0 = tm << 4, col0 = tn << 4;
  int half = lane >> 4, lr = lane & 15;
  const float* sa = qs + (size_t)(row0 + lr) * K + 2 * half;
  const float* sb = qs + (size_t)(col0 + lr) * K + 2 * half;
  const float* ta = qt + (size_t)(row0 + lr) * K + 2 * half;
  const float* tb = qt + (size_t)(col0 + lr) * K + 2 * half;
  v8f accS = {};
  v8f accT = {};
  for (int k = 0; k < K; k += 4) {
    accS = WMMA_F32(ldv2(sa + k), ldv2(sb + k), accS);
    accT = WMMA_F32(ldv2(ta + k), ldv2(tb + k), accT);
  }
  float local = 0.0f;
  int j = col0 + lr;
#pragma unroll
  for (int r = 0; r < 8; r++) {
    int i = row0 + r + 8 * half;
    if (i < Mreal && j < Mreal) {
      float gs = accS[r] * invns[i] * invns[j];
      float gt = accT[r] * invnt[i] * invnt[j];
      float d = gt - gs + 1e-6f;
      local += d * d;
    }
  }
  for (int off = 16; off >= 1; off >>= 1) local += __shfl_xor(local, off, 32);
  if (lane == 0) atomicAdd(out, local / (float)Mreal);
}

// mob loss: one wave per 16-row block; streams emb@emb^T twice (max pass +
// sumexp/mob pass).  loss += sum_i B_i*lse_i - sum_ij mob_ij*z_ij.
__global__ void mob_loss_kernel(const float* __restrict__ emb, const float* __restrict__ mob,
                                float* __restrict__ out, int N, int D) {
  int wave = (int)(((size_t)blockIdx.x * blockDim.x + threadIdx.x) >> 5);
  int lane = threadIdx.x & 31;
  int nrb = N >> 4;
  if (wave >= nrb) return;
  int row0 = wave << 4;
  int half = lane >> 4, lr = lane & 15;
  int nct = N >> 4;
  const float* arow = emb + (size_t)(row0 + lr) * D + 2 * half;

  float runmax[8];
#pragma unroll
  for (int r = 0; r < 8; r++) runmax[r] = -3.0e38f;

  // pass 1: per-row max
  for (int ct = 0; ct < nct; ct++) {
    int col0 = ct << 4;
    const float* brow = emb + (size_t)(col0 + lr) * D + 2 * half;
    v8f acc = {};
    for (int k = 0; k < D; k += 4) acc = WMMA_F32(ldv2(arow + k), ldv2(brow + k), acc);
#pragma unroll
    for (int r = 0; r < 8; r++) {
      float m = acc[r];
      m = fmaxf(m, __shfl_xor(m, 8, 32));
      m = fmaxf(m, __shfl_xor(m, 4, 32));
      m = fmaxf(m, __shfl_xor(m, 2, 32));
      m = fmaxf(m, __shfl_xor(m, 1, 32));
      runmax[r] = fmaxf(runmax[r], m);
    }
  }

  // pass 2: sumexp + mob terms
  float runsum[8], bacc[8];
  float mz = 0.0f;
#pragma unroll
  for (int r = 0; r < 8; r++) { runsum[r] = 0.0f; bacc[r] = 0.0f; }
  for (int ct = 0; ct < nct; ct++) {
    int col0 = ct << 4;
    const float* brow = emb + (size_t)(col0 + lr) * D + 2 * half;
    v8f acc = {};
    for (int k = 0; k < D; k += 4) acc = WMMA_F32(ldv2(arow + k), ldv2(brow + k), acc);
#pragma unroll
    for (int r = 0; r < 8; r++) {
      float z = acc[r];
      runsum[r] += __expf(z - runmax[r]);
      float mv = mob[(size_t)(row0 + r + 8 * half) * N + col0 + lr];
      bacc[r] += mv;
      mz += mv * z;
    }
  }
  float contrib = 0.0f;
#pragma unroll
  for (int r = 0; r < 8; r++) {
    float s = runsum[r];
    s += __shfl_xor(s, 8, 32); s += __shfl_xor(s, 4, 32);
    s += __shfl_xor(s, 2, 32); s += __shfl_xor(s, 1, 32);
    float b = bacc[r];
    b += __shfl_xor(b, 8, 32); b += __shfl_xor(b, 4, 32);
    b += __shfl_xor(b, 2, 32); b += __shfl_xor(b, 1, 32);
    contrib += b * (runmax[r] + __logf(s));
  }
  mz += __shfl_xor(mz, 16, 32); mz += __shfl_xor(mz, 8, 32); mz += __shfl_xor(mz, 4, 32);
  mz += __shfl_xor(mz, 2, 32); mz += __shfl_xor(mz, 1, 32);
  if (lr == 0) atomicAdd(out, contrib);     // lanes 0 and 16 cover rows 0-7 / 8-15
  if (lane == 0) atomicAdd(out, -mz);
}

// ---------------------------------------------------------------------------
// GAT pieces
// ---------------------------------------------------------------------------
__global__ void attn_prep_kernel(const float* __restrict__ xh, const float* __restrict__ att_src,
                                 const float* __restrict__ att_dst, float* __restrict__ a_src,
                                 float* __restrict__ a_dst, int N) {
  int idx = blockIdx.x * blockDim.x + threadIdx.x;
  if (idx >= N * 8) return;
  int n = idx >> 3, h = idx & 7;
  const float* xp = xh + (size_t)n * 1024 + h * 128;
  const float* as = att_src + h * 128;
  const float* ad = att_dst + h * 128;
  float s1 = 0.0f, s2 = 0.0f;
  for (int c = 0; c < 128; c++) {
    float x = xp[c];
    s1 += x * as[c];
    s2 += x * ad[c];
  }
  a_src[idx] = s1;
  a_dst[idx] = s2;
}

// we[h] = dot(lin_e[h*128:...], att_edge[h])  (lin_edge is [1024,1] flat)
__global__ void we_edge_kernel(const float* __restrict__ lin_e, const float* __restrict__ att_edge,
                               float* __restrict__ we) {
  __shared__ float red[128];
  int h = blockIdx.x, c = threadIdx.x;
  red[c] = lin_e[h * 128 + c] * att_edge[h * 128 + c];
  __syncthreads();
  for (int w = 64; w > 0; w >>= 1) {
    if (c < w) red[c] += red[c + w];
    __syncthreads();
  }
  if (c == 0) we[h] = red[0];
}

__global__ void edge_alpha_kernel(const int* __restrict__ src, const int* __restrict__ dst,
                                  const float* __restrict__ ew, const float* __restrict__ ewsum,
                                  const float* __restrict__ a_src, const float* __restrict__ a_dst,
                                  const float* __restrict__ we, float* __restrict__ alpha,
                                  float* __restrict__ amax, int E, int N) {
  int e = blockIdx.x * blockDim.x + threadIdx.x;
  if (e >= E + N) return;
  int s, d;
  float w;
  if (e < E) { s = src[e]; d = dst[e]; w = ew[e]; }
  else       { s = e - E;  d = e - E;  w = ewsum[0] / (float)E; }
#pragma unroll
  for (int h = 0; h < 8; h++) {
    float a = a_src[s * 8 + h] + a_dst[d * 8 + h] + w * we[h];
    a = (a >= 0.0f) ? a : 0.2f * a;
    alpha[(size_t)e * 8 + h] = a;
    atomicMaxF(&amax[d * 8 + h], a);
  }
}

__global__ void edge_accum_kernel(const int* __restrict__ src, const int* __restrict__ dst,
                                  const float* __restrict__ alpha, const float* __restrict__ amax,
                                  const float* __restrict__ xh, float* __restrict__ num,
                                  float* __restrict__ denom, int E, int N) {
  int e = blockIdx.x;
  __shared__ float exs[8];
  __shared__ int sd[2];
  if (threadIdx.x == 0) {
    sd[0] = (e < E) ? src[e] : e - E;
    sd[1] = (e < E) ? dst[e] : e - E;
  }
  __syncthreads();
  int s = sd[0], d = sd[1];
  if (threadIdx.x < 8) {
    int h = threadIdx.x;
    float ex = __expf(alpha[(size_t)e * 8 + h] - amax[d * 8 + h]);
    exs[h] = ex;
    atomicAdd(&denom[d * 8 + h], ex);
  }
  __syncthreads();
  const float* xp = xh + (size_t)s * 1024;
  float* np = num + (size_t)d * 1024;
  for (int i = threadIdx.x; i < 1024; i += 256) atomicAdd(&np[i], exs[i >> 7] * xp[i]);
}

__global__ void gat_finalize_kernel(const float* __restrict__ num, const float* __restrict__ denom,
                                    const float* __restrict__ bias, float* __restrict__ outp,
                                    int N, int doPrelu, const float* __restrict__ prelu_a) {
  int idx = blockIdx.x * blockDim.x + threadIdx.x;
  if (idx >= N * 128) return;
  int n = idx >> 7, c = idx & 127;
  float acc = 0.0f;
#pragma unroll
  for (int h = 0; h < 8; h++)
    acc += num[(size_t)n * 1024 + h * 128 + c] / (denom[n * 8 + h] + 1e-16f);
  float v = acc * 0.125f + bias[c];
  if (doPrelu) v = (v >= 0.0f) ? v : prelu_a[0] * v;
  outp[idx] = v;
}

__global__ void rownorm_inv_kernel(const float* __restrict__ X, float* __restrict__ invn, int cols) {
  __shared__ float red[256];
  int i = blockIdx.x;
  float s = 0.0f;
  for (int j = threadIdx.x; j < cols; j += blockDim.x) {
    float v = X[(size_t)i * cols + j];
    s += v * v;
  }
  red[threadIdx.x] = s;
  __syncthreads();
  for (int w = 128; w > 0; w >>= 1) {
    if ((int)threadIdx.x < w) red[threadIdx.x] += red[threadIdx.x + w];
    __syncthreads();
  }
  if (threadIdx.x == 0) invn[i] = 1.0f / fmaxf(sqrtf(red[0]), 1e-12f);
}

// ---------------------------------------------------------------------------
// host side
// ---------------------------------------------------------------------------
template <bool TRANSB>
static void launch_gemm(const float* A, const float* B, float* C, int M, int N, int K,
                        int lda, int ldb, int ldc, float alpha, const float* bias,
                        const float* resid, float rc, hipStream_t stream) {
  int waves = (M >> 4) * (N >> 4);
  int blocks = (waves * 32 + 255) / 256;
  wmma_gemm_kernel<TRANSB><<<blocks, 256, 0, stream>>>(A, B, C, M, N, K, lda, ldb, ldc,
                                                       alpha, bias, resid, rc);
}

extern "C" void kernel_launch(void* const* d_in, const int* in_sizes, int n_in,
                              void* d_out, int out_size, void* d_ws, size_t ws_size,
                              hipStream_t stream) {
  (void)n_in; (void)out_size; (void)ws_size;
  const int D = 128, H = 8;
  const int N = in_sizes[0] / D;         // 6000
  const int E = in_sizes[2] / 2;         // 192000
  const int S = in_sizes[8] / D;         // 1000
  const int Sp = ((S + 15) / 16) * 16;   // 1008
  const int EN = E + N;

  const float* s_emb = (const float*)d_in[0];
  const float* t_emb = (const float*)d_in[1];
  const int* s_eidx = (const int*)d_in[2];
  const int* t_eidx = (const int*)d_in[3];
  const float* s_ew = (const float*)d_in[4];
  const float* t_ew = (const float*)d_in[5];
  const float* mob_s = (const float*)d_in[6];
  const float* mob_t = (const float*)d_in[7];
  const float* q_emb = (const float*)d_in[8];
  const float* gat_lin[2]   = {(const float*)d_in[9],  (const float*)d_in[15]};
  const float* gat_asrc[2]  = {(const float*)d_in[10], (const float*)d_in[16]};
  const float* gat_adst[2]  = {(const float*)d_in[11], (const float*)d_in[17]};
  const float* gat_line[2]  = {(const float*)d_in[12], (const float*)d_in[18]};
  const float* gat_aedge[2] = {(const float*)d_in[13], (const float*)d_in[19]};
  const float* gat_bias[2]  = {(const float*)d_in[14], (const float*)d_in[20]};
  const float* wq_w = (const float*)d_in[21];
  const float* wq_b = (const float*)d_in[22];
  const float* wk_w = (const float*)d_in[23];
  const float* wk_b = (const float*)d_in[24];
  const float* prelu_a = (const float*)d_in[25];

  float* out = (float*)d_out;  // [s_intra, t_intra, cross_rec, cross_align]
  float* ws = (float*)d_ws;
  size_t off = 0;
  // every allocation rounded to an even float count -> 8B alignment everywhere
  auto alloc = [&](size_t nf) {
    nf = (nf + 1) & ~(size_t)1;
    float* p = ws + off;
    off += nf;
    return p;
  };
  float* xh     = alloc((size_t)N * H * D);
  float* numb   = alloc((size_t)N * H * D);
  float* denomb = alloc((size_t)N * H);
  float* amaxb  = alloc((size_t)N * H);
  float* asrcb  = alloc((size_t)N * H);
  float* adstb  = alloc((size_t)N * H);
  float* alphab = alloc((size_t)EN * H);
  float* webuf  = alloc(8);
  float* ewsum  = alloc(2);
  float* cityA  = alloc((size_t)N * D);
  float* sOut   = alloc((size_t)N * D);
  float* tOut   = alloc((size_t)N * D);
  float* qpad   = alloc((size_t)Sp * D);
  float* qsb    = alloc((size_t)Sp * N);
  float* qtb    = alloc((size_t)Sp * N);
  float* invns  = alloc(Sp);
  float* invnt  = alloc(Sp);
  float* sq     = alloc((size_t)N * D);
  float* skb    = alloc((size_t)N * D);
  float* tq     = alloc((size_t)N * D);
  float* tkb    = alloc((size_t)N * D);
  float* Mt     = alloc((size_t)D * D);
  float* Ms     = alloc((size_t)D * D);
  float* Us     = alloc((size_t)N * D);
  float* Ut     = alloc((size_t)N * D);

  auto fill = [&](float* p, float v, size_t n) {
    fill_kernel<<<(unsigned)((n + 255) / 256), 256, 0, stream>>>(p, v, (int)n);
  };

  fill(out, 0.0f, 4);

  // ------------------------------ GAT layers ------------------------------
  auto run_gat_layer = [&](const float* xin, float* xout, int city, int layer,
                           const int* eidx, const float* ew, bool doPrelu) {
    const float* linw = gat_lin[city] + (size_t)layer * H * D * D;
    const float* as   = gat_asrc[city] + (size_t)layer * H * D;
    const float* ad   = gat_adst[city] + (size_t)layer * H * D;
    const float* le   = gat_line[city] + (size_t)layer * H * D;
    const float* ae   = gat_aedge[city] + (size_t)layer * H * D;
    const float* bs   = gat_bias[city] + (size_t)layer * D;
    // xh = x @ lin_w^T   [N, H*D]
    launch_gemm<true>(xin, linw, xh, N, H * D, D, D, D, H * D, 1.0f, nullptr, nullptr, 0.0f, stream);
    attn_prep_kernel<<<(N * H + 255) / 256, 256, 0, stream>>>(xh, as, ad, asrcb, adstb, N);
    we_edge_kernel<<<H, 128, 0, stream>>>(le, ae, webuf);
    fill(ewsum, 0.0f, 1);
    reduce_sum_kernel<<<256, 256, 0, stream>>>(ew, E, ewsum);
    fill(amaxb, -3.0e38f, (size_t)N * H);
    fill(denomb, 0.0f, (size_t)N * H);
    fill(numb, 0.0f, (size_t)N * H * D);
    edge_alpha_kernel<<<(EN + 255) / 256, 256, 0, stream>>>(eidx, eidx + E, ew, ewsum,
                                                            asrcb, adstb, webuf, alphab, amaxb, E, N);
    edge_accum_kernel<<<EN, 256, 0, stream>>>(eidx, eidx + E, alphab, amaxb, xh, numb, denomb, E, N);
    gat_finalize_kernel<<<(N * D + 255) / 256, 256, 0, stream>>>(numb, denomb, bs, xout, N,
                                                                 doPrelu ? 1 : 0, prelu_a);
  };

  run_gat_layer(s_emb, cityA, 0, 0, s_eidx, s_ew, true);
  run_gat_layer(cityA, sOut, 0, 1, s_eidx, s_ew, false);
  run_gat_layer(t_emb, cityA, 1, 0, t_eidx, t_ew, true);
  run_gat_layer(cityA, tOut, 1, 1, t_eidx, t_ew, false);

  // ------------------------------ mob losses ------------------------------
  {
    int waves = N / 16;
    int blocks = (waves * 32 + 255) / 256;
    mob_loss_kernel<<<blocks, 256, 0, stream>>>(sOut, mob_s, out + 0, N, D);
    mob_loss_kernel<<<blocks, 256, 0, stream>>>(tOut, mob_t, out + 1, N, D);
  }

  // ------------------------------ cross_align ------------------------------
  pad_copy_kernel<<<(Sp * D + 255) / 256, 256, 0, stream>>>(q_emb, qpad, S * D, Sp * D);
  launch_gemm<true>(qpad, sOut, qsb, Sp, N, D, D, D, N, 1.0f, nullptr, nullptr, 0.0f, stream);
  launch_gemm<true>(qpad, tOut, qtb, Sp, N, D, D, D, N, 1.0f, nullptr, nullptr, 0.0f, stream);
  rownorm_inv_kernel<<<Sp, 256, 0, stream>>>(qsb, invns, N);
  rownorm_inv_kernel<<<Sp, 256, 0, stream>>>(qtb, invnt, N);
  {
    int waves = (Sp / 16) * (Sp / 16);
    int blocks = (waves * 32 + 255) / 256;
    align_loss_kernel<<<blocks, 256, 0, stream>>>(qsb, qtb, invns, invnt, out + 3, Sp, S, N);
  }

  // ------------------------------ cross_rec ------------------------------
  // st@ts = s_q (scale^2 t_k^T t_q) s_k^T  =>  ds = (s_q@Mt - scale*s_q) @ s_k^T + eps
  float scale = 1.0f / sqrtf((float)D);
  launch_gemm<true>(sOut, wq_w, sq, N, D, D, D, D, D, 1.0f, wq_b, nullptr, 0.0f, stream);
  launch_gemm<true>(sOut, wk_w, skb, N, D, D, D, D, D, 1.0f, wk_b, nullptr, 0.0f, stream);
  launch_gemm<true>(tOut, wq_w, tq, N, D, D, D, D, D, 1.0f, wq_b, nullptr, 0.0f, stream);
  launch_gemm<true>(tOut, wk_w, tkb, N, D, D, D, D, D, 1.0f, wk_b, nullptr, 0.0f, stream);
  {
    int waves = (D / 16) * (D / 16);
    int blocks = (waves * 32 + 255) / 256;
    wmma_gemm_tn_kernel<<<blocks, 256, 0, stream>>>(tkb, tq, Mt, D, D, N, D, D, D, scale * scale);
    wmma_gemm_tn_kernel<<<blocks, 256, 0, stream>>>(skb, sq, Ms, D, D, N, D, D, D, scale * scale);
  }
  launch_gemm<false>(sq, Mt, Us, N, D, D, D, D, D, 1.0f, nullptr, sq, -scale, stream);
  launch_gemm<false>(tq, Ms, Ut, N, D, D, D, D, D, 1.0f, nullptr, tq, -scale, stream);
  {
    int waves = (N / 16) * (N / 16);
    int blocks = (waves * 32 + 255) / 256;
    rec_loss_kernel<<<blocks, 256, 0, stream>>>(Us, skb, out + 2, N, N, D, 1.0f / (float)N);
    rec_loss_kernel<<<blocks, 256, 0, stream>>>(Ut, tkb, out + 2, N, N, D, 1.0f / (float)N);
  }
}